// SuperGAT_15556371546773
// MI455X (gfx1250) — compile-verified
//
#include <hip/hip_runtime.h>
#include <math.h>

typedef __attribute__((ext_vector_type(2))) float v2f;
typedef __attribute__((ext_vector_type(8))) float v8f;

#define NNODES 50000
#define NEDGES 640000
#define ETOT   (NEDGES + NNODES)
#define NEG_SLOPE 0.2f

// -------- float atomic max via integer atomics (works with init = -inf) -----
__device__ __forceinline__ void atomicMaxFloat(float* addr, float val) {
    if (val >= 0.0f)
        atomicMax((int*)addr, __float_as_int(val));
    else
        atomicMin((unsigned int*)addr, __float_as_uint(val));
}

// ---------------- GEMM: out[nrows x NCOL] = in[nrows x K] * W[K x NCOL] -----
// fp32 WMMA 16x16x4.  One block = 16 rows, one wave per 16-col tile.
// NNODES % 16 == 0, so no predication (EXEC all-1s as WMMA requires).
template<int K, int NCOL>
__global__ __launch_bounds__((NCOL / 16) * 32)
void gemm_wmma_f32(const float* __restrict__ in, const float* __restrict__ W,
                   float* __restrict__ out) {
    const int lane = threadIdx.x & 31;
    const int wave = threadIdx.x >> 5;
    const int hp   = lane >> 4;   // lane-half: 0 or 1
    const int lm   = lane & 15;
    const int row0 = blockIdx.x * 16;
    const int c0   = wave * 16;

    v8f acc = {};
    const float* arow = in + (size_t)(row0 + lm) * K;
    #pragma unroll
    for (int k0 = 0; k0 < K; k0 += 4) {
        // A 16x4: lane holds row M=lm; VGPR0/1 = K{0,1} (hp=0) or K{2,3} (hp=1)
        v2f a;
        a.x = arow[k0 + hp * 2 + 0];
        a.y = arow[k0 + hp * 2 + 1];
        // B 4x16: lane holds col N=c0+lm; same K striping as A
        v2f b;
        b.x = W[(size_t)(k0 + hp * 2 + 0) * NCOL + c0 + lm];
        b.y = W[(size_t)(k0 + hp * 2 + 1) * NCOL + c0 + lm];
        acc = __builtin_amdgcn_wmma_f32_16x16x4_f32(
            /*neg_a=*/false, a, /*neg_b=*/false, b,
            /*c_mod=*/(short)0, acc, /*reuse_a=*/false, /*reuse_b=*/false);
    }
    // C/D layout: VGPR v -> row M = v + hp*8, col N = lm
    #pragma unroll
    for (int v = 0; v < 8; ++v)
        out[(size_t)(row0 + v + hp * 8) * NCOL + c0 + lm] = acc[v];
}

// ---- per-layer init: zero aggregation buffer, amax=-inf, denom=0 -----------
__global__ void init_layer(float* __restrict__ outbuf, int out_len,
                           float* __restrict__ amax, float* __restrict__ denom) {
    int i = blockIdx.x * blockDim.x + threadIdx.x;
    if (i < out_len) outbuf[i] = 0.0f;
    if (i < NNODES) { amax[i] = -INFINITY; denom[i] = 0.0f; }
}

// ---- edge attention scores: one wave per edge ------------------------------
template<int C>
__global__ void edge_alpha(const float* __restrict__ h,
                           const int* __restrict__ ei,
                           const float* __restrict__ al,
                           const float* __restrict__ ar,
                           float* __restrict__ alpha,
                           float* __restrict__ amax) {
    const int lane = threadIdx.x & 31;
    const int e = blockIdx.x * (blockDim.x >> 5) + (threadIdx.x >> 5);
    if (e >= ETOT) return;
    int s, d;
    if (e < NEDGES) { s = ei[e]; d = ei[NEDGES + e]; }
    else            { s = e - NEDGES; d = s; }

    float dot = 0.0f, sl = 0.0f, sr = 0.0f;
    #pragma unroll
    for (int t = 0; t < C / 32; ++t) {
        int c = t * 32 + lane;                       // coalesced across lanes
        float hj = h[(size_t)s * C + c];
        float hi = h[(size_t)d * C + c];
        dot = fmaf(hj, hi, dot);
        sl  = fmaf(hj, al[c], sl);
        sr  = fmaf(hi, ar[c], sr);
    }
    #pragma unroll
    for (int off = 16; off > 0; off >>= 1) {
        dot += __shfl_xor(dot, off, 32);
        sl  += __shfl_xor(sl,  off, 32);
        sr  += __shfl_xor(sr,  off, 32);
    }
    if (lane == 0) {
        float sig = 1.0f / (1.0f + expf(-dot));      // MX gate
        float a = (sl + sr) * sig;                   // GO score * gate
        a = (a >= 0.0f) ? a : NEG_SLOPE * a;         // leaky_relu
        alpha[e] = a;
        atomicMaxFloat(&amax[d], a);                 // segment max
    }
}

// ---- exp and denom accumulation: one thread per edge -----------------------
__global__ void edge_expsum(const int* __restrict__ ei,
                            float* __restrict__ alpha,
                            const float* __restrict__ amax,
                            float* __restrict__ denom) {
    int e = blockIdx.x * blockDim.x + threadIdx.x;
    if (e >= ETOT) return;
    int d = (e < NEDGES) ? ei[NEDGES + e] : (e - NEDGES);
    float ex = expf(alpha[e] - amax[d]);
    alpha[e] = ex;                                   // overwrite in place
    atomicAdd(&denom[d], ex);
}

// ---- weighted scatter-aggregation: one wave per edge -----------------------
template<int C>
__global__ void edge_aggregate(const float* __restrict__ h,
                               const int* __restrict__ ei,
                               const float* __restrict__ alpha,
                               const float* __restrict__ denom,
                               float* __restrict__ out) {
    const int lane = threadIdx.x & 31;
    const int e = blockIdx.x * (blockDim.x >> 5) + (threadIdx.x >> 5);
    if (e >= ETOT) return;
    int s, d;
    if (e < NEDGES) { s = ei[e]; d = ei[NEDGES + e]; }
    else            { s = e - NEDGES; d = s; }
    float w = alpha[e] / (denom[d] + 1e-16f);        // uniform across wave
    #pragma unroll
    for (int t = 0; t < C / 32; ++t) {
        int c = t * 32 + lane;
        atomicAdd(&out[(size_t)d * C + c], h[(size_t)s * C + c] * w);
    }
}

// ---- bias + relu (in place) ------------------------------------------------
__global__ void bias_relu(float* __restrict__ out, const float* __restrict__ b,
                          int total, int cmask) {
    int i = blockIdx.x * blockDim.x + threadIdx.x;
    if (i >= total) return;
    float v = out[i] + b[i & cmask];
    out[i] = v > 0.0f ? v : 0.0f;
}

// ---- log_softmax over 64 cols: one wave per node ---------------------------
__global__ void log_softmax64(const float* __restrict__ in, float* __restrict__ out) {
    const int lane = threadIdx.x & 31;
    const int n = blockIdx.x * (blockDim.x >> 5) + (threadIdx.x >> 5);
    if (n >= NNODES) return;
    float v0 = in[(size_t)n * 64 + lane];
    float v1 = in[(size_t)n * 64 + 32 + lane];
    float m = fmaxf(v0, v1);
    #pragma unroll
    for (int off = 16; off > 0; off >>= 1)
        m = fmaxf(m, __shfl_xor(m, off, 32));
    float s = expf(v0 - m) + expf(v1 - m);
    #pragma unroll
    for (int off = 16; off > 0; off >>= 1)
        s += __shfl_xor(s, off, 32);
    float ls = m + logf(s);
    out[(size_t)n * 64 + lane]      = v0 - ls;
    out[(size_t)n * 64 + 32 + lane] = v1 - ls;
}

// ---------------- layer driver (host) ---------------------------------------
template<int K, int C>
static void run_layer(const float* in, const float* W, const float* al,
                      const float* ar, const float* b, const int* ei,
                      float* hbuf, float* outbuf, float* alpha,
                      float* amax, float* denom, hipStream_t stream) {
    // GEMM first: it is the only consumer of `in` (which may alias outbuf
    // from the previous layer); init zeroes outbuf afterwards.
    gemm_wmma_f32<K, C><<<NNODES / 16, (C / 16) * 32, 0, stream>>>(in, W, hbuf);
    const int out_len = NNODES * C;
    init_layer<<<(out_len + 255) / 256, 256, 0, stream>>>(outbuf, out_len, amax, denom);
    const int WPB = 8;                         // waves per block
    const int gridE = (ETOT + WPB - 1) / WPB;
    edge_alpha<C><<<gridE, WPB * 32, 0, stream>>>(hbuf, ei, al, ar, alpha, amax);
    edge_expsum<<<(ETOT + 255) / 256, 256, 0, stream>>>(ei, alpha, amax, denom);
    edge_aggregate<C><<<gridE, WPB * 32, 0, stream>>>(hbuf, ei, alpha, denom, outbuf);
    bias_relu<<<(out_len + 255) / 256, 256, 0, stream>>>(outbuf, b, out_len, C - 1);
}

extern "C" void kernel_launch(void* const* d_in, const int* in_sizes, int n_in,
                              void* d_out, int out_size, void* d_ws, size_t ws_size,
                              hipStream_t stream) {
    const float* x   = (const float*)d_in[0];
    const int*   ei  = (const int*)  d_in[1];   // [2, E] row-major
    const float* W0  = (const float*)d_in[2];
    const float* al0 = (const float*)d_in[3];
    const float* ar0 = (const float*)d_in[4];
    const float* b0  = (const float*)d_in[5];
    const float* W1  = (const float*)d_in[6];
    const float* al1 = (const float*)d_in[7];
    const float* ar1 = (const float*)d_in[8];
    const float* b1  = (const float*)d_in[9];
    const float* W2  = (const float*)d_in[10];
    const float* al2 = (const float*)d_in[11];
    const float* ar2 = (const float*)d_in[12];
    const float* b2  = (const float*)d_in[13];

    float* ws     = (float*)d_ws;
    float* hbuf   = ws;                                   // N*128
    float* outbuf = hbuf   + (size_t)NNODES * 128;        // N*128
    float* alpha  = outbuf + (size_t)NNODES * 128;        // ETOT
    float* amax   = alpha  + (size_t)ETOT;                // N
    float* denom  = amax   + (size_t)NNODES;              // N

    run_layer<128, 128>(x,      W0, al0, ar0, b0, ei, hbuf, outbuf, alpha, amax, denom, stream);
    run_layer<128, 128>(outbuf, W1, al1, ar1, b1, ei, hbuf, outbuf, alpha, amax, denom, stream);
    run_layer<128,  64>(outbuf, W2, al2, ar2, b2, ei, hbuf, outbuf, alpha, amax, denom, stream);

    const int WPB = 8;
    log_softmax64<<<(NNODES + WPB - 1) / WPB, WPB * 32, 0, stream>>>(outbuf, (float*)d_out);
}